// HyperbolicAttention_72705206386750
// MI455X (gfx1250) — compile-verified
//
#include <hip/hip_runtime.h>
#include <math.h>

// ---------------------------------------------------------------------------
// Hyperbolic multi-head attention for MI455X (gfx1250), fp32 WMMA path.
//   bsz=2, seq=512, dim=512, heads=8, hd=64, c=1
// Distance reduces to a function of (|q|^2, |k|^2, <q,k>) so everything heavy
// is a GEMM -> V_WMMA_F32_16X16X4_F32 keeps fp32 accuracy vs the reference.
// ---------------------------------------------------------------------------

typedef __attribute__((ext_vector_type(2))) float v2f;
typedef __attribute__((ext_vector_type(8))) float v8f;

#define WMMA_F32(a, b, c) \
  __builtin_amdgcn_wmma_f32_16x16x4_f32(false, (a), false, (b), (short)0, (c), false, false)

#define DIMSZ 512
#define SEQSZ 512
#define BSZ   2
#define NTOK  (BSZ * SEQSZ)
#define NHEAD 8
#define HDIM  64
#define PSTRIDE 516   // 16x516 f32 LDS tile: bank-conflict-free fragment reads

static __device__ __forceinline__ v2f ld2(const float* p) {
  const float2 t = *reinterpret_cast<const float2*>(p);
  v2f r; r.x = t.x; r.y = t.y; return r;
}

// Y[m][n] = sum_k X[m][k] * W[n][k] + bias[n]   (i.e. X @ W^T + b)
// one wave per 16x16 output tile; K accumulated 4 at a time via f32 WMMA.
__global__ __launch_bounds__(32) void gemm_xwt_bias(
    const float* __restrict__ X, const float* __restrict__ W,
    const float* __restrict__ bias, float* __restrict__ Y) {
  const int m0   = blockIdx.x * 16;
  const int n0   = blockIdx.y * 16;
  const int lane = threadIdx.x & 31;
  const int lm   = lane & 15;
  const int half = lane >> 4;

  const float bn = bias[n0 + lm];
  v8f acc;
#pragma unroll
  for (int r = 0; r < 8; ++r) acc[r] = bn;  // bias is per-column: same for all rows

  const float* xrow = X + (size_t)(m0 + lm) * DIMSZ;
  const float* wrow = W + (size_t)(n0 + lm) * DIMSZ;
#pragma unroll 8
  for (int k0 = 0; k0 < DIMSZ; k0 += 4) {
    v2f a = ld2(xrow + k0 + 2 * half);  // A: lane<16 holds K={k0,k0+1}, lane>=16 K={k0+2,k0+3}
    v2f b = ld2(wrow + k0 + 2 * half);  // B[k][n] = W[n][k], column-major per lane
    acc = WMMA_F32(a, b, acc);
  }
#pragma unroll
  for (int r = 0; r < 8; ++r)
    Y[(size_t)(m0 + r + 8 * half) * DIMSZ + n0 + lm] = acc[r];
}

// In-place exp_map_zero (with projection) per (token, head) 64-dim row.
// Also writes the post-map squared norm (x2) used by the distance formula.
__global__ __launch_bounds__(256) void expmap_rows(float* __restrict__ QK,
                                                   float* __restrict__ sq) {
  const int row  = blockIdx.x * 8 + (threadIdx.x >> 5);  // row = tok*8 + h
  const int lane = threadIdx.x & 31;
  const int tok  = row >> 3;
  const int h    = row & 7;
  float* base = QK + (size_t)tok * DIMSZ + h * HDIM;

  float e0 = base[lane];
  float e1 = base[lane + 32];
  float ss = e0 * e0 + e1 * e1;
#pragma unroll
  for (int off = 16; off >= 1; off >>= 1) ss += __shfl_xor(ss, off, 32);

  const float nrm    = sqrtf(ss);
  const float vn     = fmaxf(nrm, 1e-5f);
  const float factor = tanhf(vn) / vn;           // tanh(sqrt_c*|v|)/(sqrt_c*|v|), c=1
  const float wn     = factor * nrm;             // norm after exp map
  const float ps     = fminf(0.999f / fmaxf(wn, 1e-5f), 1.0f);  // _project clamp
  const float total  = factor * ps;

  base[lane]      = e0 * total;
  base[lane + 32] = e1 * total;
  if (lane == 0) {
    const float pn = wn * ps;
    sq[row] = pn * pn;
  }
}

// One workgroup = 4 waves = 16 query rows of one (batch, head).
// Stage 1: wave w computes S = Qh Kh^T over keys [w*128, w*128+128) (WMMA, K=64)
//          -> closed-form hyperbolic distance -> p = exp(clip(-d/8,-50,0)) -> LDS
// Stage 2: LDS row sums; wave w computes O(:, w*16..+16) = P V (WMMA, K=512),
//          scaling rows by 1/rowsum at the store.
__global__ __launch_bounds__(128) void hyp_attn(
    const float* __restrict__ Qs, const float* __restrict__ Ks,
    const float* __restrict__ V,  const float* __restrict__ q2,
    const float* __restrict__ k2, float* __restrict__ AO) {
  __shared__ float P[16 * PSTRIDE];
  __shared__ float part[16][8];
  __shared__ float rinv[16];

  const int q0   = blockIdx.x * 16;
  const int h    = blockIdx.y;
  const int b    = blockIdx.z;
  const int wv   = threadIdx.x >> 5;
  const int lane = threadIdx.x & 31;
  const int lm   = lane & 15;
  const int half = lane >> 4;
  const int tokq = b * SEQSZ + q0;   // first query token of this block
  const int tokb = b * SEQSZ;        // first key token of this batch
  const int hbase = h * HDIM;

  // ---- Stage 1: S tiles -------------------------------------------------
  v8f acc[8];
#pragma unroll
  for (int t = 0; t < 8; ++t)
#pragma unroll
    for (int r = 0; r < 8; ++r) acc[t][r] = 0.0f;

  const float* qrow = Qs + (size_t)(tokq + lm) * DIMSZ + hbase;
#pragma unroll 4
  for (int ks = 0; ks < HDIM; ks += 4) {
    v2f a = ld2(qrow + ks + 2 * half);  // shared across all 8 key tiles
#pragma unroll
    for (int t = 0; t < 8; ++t) {
      const int n0 = wv * 128 + t * 16;
      v2f bb = ld2(Ks + (size_t)(tokb + n0 + lm) * DIMSZ + hbase + ks + 2 * half);
      acc[t] = WMMA_F32(a, bb, acc[t]);
    }
  }

  // per-row x2 (depends only on local row lr = r + 8*half)
  float x2v[8];
#pragma unroll
  for (int r = 0; r < 8; ++r)
    x2v[r] = q2[(size_t)(tokq + r + 8 * half) * NHEAD + h];

#pragma unroll
  for (int t = 0; t < 8; ++t) {
    const int n = wv * 128 + t * 16 + lm;
    const float y2 = k2[(size_t)(tokb + n) * NHEAD + h];
#pragma unroll
    for (int r = 0; r < 8; ++r) {
      const int lr  = r + 8 * half;
      const float x2 = x2v[r];
      const float xy = acc[t][r];
      // ||(-x) (+) y|| in closed form (c = 1)
      const float A   = 1.0f - 2.0f * xy + y2;
      const float Bc  = 1.0f - x2;
      const float den = fmaxf(1.0f - 2.0f * xy + x2 * y2, 1e-5f);
      float num2 = A * A * x2 + Bc * Bc * y2 - 2.0f * A * Bc * xy;
      num2 = fmaxf(num2, 0.0f);
      const float nrm = sqrtf(num2) / den;
      const float z   = fminf(nrm, 0.999f);                 // project + distance clamps
      const float d   = logf((1.0f + z) / (1.0f - z));      // 2*arctanh(z)
      const float lg  = fmaxf(fminf(-d * 0.125f, 0.0f), -50.0f);
      P[lr * PSTRIDE + n] = expf(lg);
    }
  }
  __syncthreads();

  // ---- Row sums ----------------------------------------------------------
  {
    const int row = threadIdx.x >> 3;  // 16 rows x 8 threads
    const int ch  = threadIdx.x & 7;
    float s = 0.0f;
#pragma unroll 8
    for (int i = 0; i < 64; ++i) s += P[row * PSTRIDE + ch + i * 8];  // strided: no 8-way bank conflict
    part[row][ch] = s;
  }
  __syncthreads();
  if (threadIdx.x < 16) {
    float s = 0.0f;
#pragma unroll
    for (int i = 0; i < 8; ++i) s += part[threadIdx.x][i];
    rinv[threadIdx.x] = 1.0f / s;
  }
  __syncthreads();

  // ---- Stage 2: O = softmax(P) * V ---------------------------------------
  v8f o;
#pragma unroll
  for (int r = 0; r < 8; ++r) o[r] = 0.0f;

  const int dc = hbase + wv * 16 + lm;  // output column (within DIM)
#pragma unroll 8
  for (int k0 = 0; k0 < SEQSZ; k0 += 4) {
    v2f a = ld2(&P[lm * PSTRIDE + k0 + 2 * half]);  // P as A-matrix from LDS
    v2f bb;
    bb.x = V[(size_t)(tokb + k0 + 2 * half) * DIMSZ + dc];
    bb.y = V[(size_t)(tokb + k0 + 2 * half + 1) * DIMSZ + dc];
    o = WMMA_F32(a, bb, o);
  }
#pragma unroll
  for (int r = 0; r < 8; ++r) {
    const int lr = r + 8 * half;
    AO[(size_t)(tokq + lr) * DIMSZ + dc] = o[r] * rinv[lr];
  }
}

extern "C" void kernel_launch(void* const* d_in, const int* in_sizes, int n_in,
                              void* d_out, int out_size, void* d_ws, size_t ws_size,
                              hipStream_t stream) {
  (void)in_sizes; (void)n_in; (void)out_size; (void)ws_size;
  const float* x  = (const float*)d_in[0];
  const float* Wq = (const float*)d_in[1];
  const float* bq = (const float*)d_in[2];
  const float* Wk = (const float*)d_in[3];
  const float* bk = (const float*)d_in[4];
  const float* Wv = (const float*)d_in[5];
  const float* bv = (const float*)d_in[6];
  const float* Wo = (const float*)d_in[7];
  const float* bo = (const float*)d_in[8];
  float* out = (float*)d_out;

  // workspace layout (floats): Q | K | V | AO | q2 | k2   (~8.1 MB)
  float* ws = (float*)d_ws;
  const size_t MAT = (size_t)NTOK * DIMSZ;  // 524288
  float* Q  = ws;
  float* K  = Q + MAT;
  float* V  = K + MAT;
  float* AO = V + MAT;
  float* q2 = AO + MAT;
  float* k2 = q2 + (size_t)NTOK * NHEAD;

  const dim3 gGemm(NTOK / 16, DIMSZ / 16);  // (64, 32), one wave per tile
  gemm_xwt_bias<<<gGemm, 32, 0, stream>>>(x, Wq, bq, Q);
  gemm_xwt_bias<<<gGemm, 32, 0, stream>>>(x, Wk, bk, K);
  gemm_xwt_bias<<<gGemm, 32, 0, stream>>>(x, Wv, bv, V);

  expmap_rows<<<NTOK * NHEAD / 8, 256, 0, stream>>>(Q, q2);
  expmap_rows<<<NTOK * NHEAD / 8, 256, 0, stream>>>(K, k2);

  hyp_attn<<<dim3(SEQSZ / 16, NHEAD, BSZ), 128, 0, stream>>>(Q, K, V, q2, k2, AO);

  gemm_xwt_bias<<<gGemm, 32, 0, stream>>>(AO, Wo, bo, out);
}